// L_spa_49898930045541
// MI455X (gfx1250) — compile-verified
//
#include <hip/hip_runtime.h>

typedef float v2f __attribute__((ext_vector_type(2)));
typedef float v4f __attribute__((ext_vector_type(4)));
typedef float v8f __attribute__((ext_vector_type(8)));

// ---------------------------------------------------------------------------
// Stage 1: fused channel-mean + 4x4 avg-pool of (org - enhance).
//   org/enh : [32,3,512,512] f32   (201 MB total -> HBM-bound, ~8.6us floor)
//   diff    : [32,128,128]  f32   (= sum over 3ch x 4x4 / 48)
// One thread per pooled output pixel; 24 coalesced nontemporal 128-bit loads.
// Lane l of a wave reads v4f index j=l within a 512-float row -> each wave
// load instruction covers 512 contiguous bytes. Nontemporal hint keeps the
// 201 MB stream from evicting the 2 MB stage-1 output from L2 (192 MB).
// ---------------------------------------------------------------------------
__global__ __launch_bounds__(256) void pool_diff_kernel(
    const v4f* __restrict__ org, const v4f* __restrict__ enh,
    float* __restrict__ diff) {
  const int idx = blockIdx.x * blockDim.x + threadIdx.x;  // 0 .. 524287
  const int j = idx & 127;          // pooled x  (== v4f column index)
  const int i = (idx >> 7) & 127;   // pooled y
  const int b = idx >> 14;          // batch
  float s = 0.0f;
#pragma unroll
  for (int c = 0; c < 3; ++c) {
#pragma unroll
    for (int r = 0; r < 4; ++r) {
      const size_t off =
          ((size_t)(b * 3 + c) * 512 + (size_t)(4 * i + r)) * 128 + j;
      const v4f o = __builtin_nontemporal_load(&org[off]);
      const v4f e = __builtin_nontemporal_load(&enh[off]);
      s += (o.x - e.x) + (o.y - e.y) + (o.z - e.z) + (o.w - e.w);
    }
  }
  diff[idx] = s * (1.0f / 48.0f);
}

// ---------------------------------------------------------------------------
// Stage 2: shift-difference energy + global mean.
//   e[p] = (D-Dl)^2 + (D-Dr)^2 + (D-Du)^2 + (D-Dd)^2   (zero padded)
//   out  = sum(e) / 524288
// Each wave handles 256 pixels (8 per lane).  The wave-level reduction is done
// on the matrix pipe: V_WMMA_F32_16X16X4_F32 with B = ones(4x16) computes
// D[m,n] = sum_k A[m,k] + C[m,n], i.e. each WMMA folds 64 f32 values (2/lane)
// into the accumulator exactly (f32 FMA path, RNE).  After 4 chained WMMAs the
// accumulator holds row totals replicated across N; per-lane sum of its 8 rows
// + one xor-16 shuffle yields the wave total in every lane.
// ---------------------------------------------------------------------------
__global__ __launch_bounds__(256) void grad_wmma_reduce_kernel(
    const float* __restrict__ D, float* __restrict__ out) {
  const int lane = threadIdx.x & 31;
  const int wave = (blockIdx.x * blockDim.x + threadIdx.x) >> 5;  // 0..2047

  const v2f ones = {1.0f, 1.0f};
  v8f acc = {0.f, 0.f, 0.f, 0.f, 0.f, 0.f, 0.f, 0.f};

#pragma unroll
  for (int t = 0; t < 4; ++t) {
    float e2[2];
#pragma unroll
    for (int k = 0; k < 2; ++k) {
      const int p = wave * 256 + t * 64 + k * 32 + lane;  // flat pixel index
      const int j = p & 127;
      const int i = (p >> 7) & 127;
      const float c  = D[p];
      const float xl = (j > 0)   ? D[p - 1]   : 0.0f;
      const float xr = (j < 127) ? D[p + 1]   : 0.0f;
      const float xu = (i > 0)   ? D[p - 128] : 0.0f;
      const float xd = (i < 127) ? D[p + 128] : 0.0f;
      const float dl = c - xl, dr = c - xr, du = c - xu, dd = c - xd;
      e2[k] = dl * dl + dr * dr + du * du + dd * dd;
    }
    v2f a = {e2[0], e2[1]};
    // D = A(16x4) * ones(4x16) + C  -> accumulate 64 values into C rows
    acc = __builtin_amdgcn_wmma_f32_16x16x4_f32(
        /*neg_a=*/false, a, /*neg_b=*/false, ones,
        /*c_mod=*/(short)0, acc, /*reuse_a=*/false, /*reuse_b=*/false);
  }

  // Lanes 0-15 hold rows 0-7, lanes 16-31 hold rows 8-15 (replicated over N).
  float s = acc[0] + acc[1] + acc[2] + acc[3] +
            acc[4] + acc[5] + acc[6] + acc[7];
  s += __shfl_xor(s, 16, 32);  // SA + SB = full wave total (all lanes)
  if (lane == 0) atomicAdd(out, s * (1.0f / 524288.0f));
}

extern "C" void kernel_launch(void* const* d_in, const int* in_sizes, int n_in,
                              void* d_out, int out_size, void* d_ws,
                              size_t ws_size, hipStream_t stream) {
  const v4f* org = (const v4f*)d_in[0];
  const v4f* enh = (const v4f*)d_in[1];
  float* out = (float*)d_out;
  float* diff = (float*)d_ws;  // 524288 floats = 2 MB scratch

  (void)hipMemsetAsync(out, 0, sizeof(float), stream);

  // 32*128*128 pooled pixels, one per thread.
  pool_diff_kernel<<<2048, 256, 0, stream>>>(org, enh, diff);

  // 2048 waves x 256 pixels each.
  grad_wmma_reduce_kernel<<<256, 256, 0, stream>>>(diff, out);
}